// DWT_Transform_43722767073762
// MI455X (gfx1250) — compile-verified
//
#include <hip/hip_runtime.h>

// Batched 2D Haar DWT (pywt 'haar' dwt2 semantics), f32 in / f32 out.
// Input : x [N, 256, 256] with N = B*C images (16*64 = 1024)
// Output: LL | LH | HL | HH concatenated, each [N, 128, 128]
//
// Pure streaming kernel: 512 MB compulsory traffic @ 23.3 TB/s -> ~22 us floor.
// Each thread: 4x global_load_b128 (NT) + 4x global_store_b128 (NT),
// fully coalesced 512B-per-wave bursts under wave32.

typedef float v4f __attribute__((ext_vector_type(4)));

__global__ __launch_bounds__(256) void haar_dwt2_kernel(
    const float* __restrict__ x,
    float* __restrict__ out,
    int n_img,
    long plane_out)   // = (long)n_img * 128 * 128
{
    const long tid = (long)blockIdx.x * (long)blockDim.x + (long)threadIdx.x;
    // 32 threads cover one output row of 128 (4 outputs each).
    const int  quad = (int)(tid & 31);          // which group of 4 output cols
    const int  row  = (int)((tid >> 5) & 127);  // output row
    const long img  = tid >> 12;                // image index
    if (img >= (long)n_img) return;

    // Input: rows 2*row and 2*row+1, columns [8*quad, 8*quad+8)
    const long in_base = img * (long)(256 * 256) + (long)(2 * row) * 256 + ((long)quad << 3);

    const v4f r0a = __builtin_nontemporal_load((const v4f*)(x + in_base));            // row even, cols 0..3
    const v4f r0b = __builtin_nontemporal_load((const v4f*)(x + in_base + 4));        // row even, cols 4..7
    const v4f r1a = __builtin_nontemporal_load((const v4f*)(x + in_base + 256));      // row odd,  cols 0..3
    const v4f r1b = __builtin_nontemporal_load((const v4f*)(x + in_base + 256 + 4));  // row odd,  cols 4..7

    v4f ll, lh, hl, hh;
#pragma unroll
    for (int j = 0; j < 2; ++j) {
        const float a = r0a[2 * j];      // even row, even col
        const float b = r0a[2 * j + 1];  // even row, odd col
        const float c = r1a[2 * j];      // odd row,  even col
        const float d = r1a[2 * j + 1];  // odd row,  odd col
        const float s = a + b, t = c + d, u = a - b, v = c - d;
        ll[j] = (s + t) * 0.5f;
        lh[j] = (s - t) * 0.5f;
        hl[j] = (u + v) * 0.5f;
        hh[j] = (u - v) * 0.5f;
    }
#pragma unroll
    for (int j = 0; j < 2; ++j) {
        const float a = r0b[2 * j];
        const float b = r0b[2 * j + 1];
        const float c = r1b[2 * j];
        const float d = r1b[2 * j + 1];
        const float s = a + b, t = c + d, u = a - b, v = c - d;
        ll[2 + j] = (s + t) * 0.5f;
        lh[2 + j] = (s - t) * 0.5f;
        hl[2 + j] = (u + v) * 0.5f;
        hh[2 + j] = (u - v) * 0.5f;
    }

    // Output: 4 consecutive floats per plane, planes laid out back-to-back.
    const long o = img * (long)(128 * 128) + (long)row * 128 + ((long)quad << 2);
    __builtin_nontemporal_store(ll, (v4f*)(out + o));
    __builtin_nontemporal_store(lh, (v4f*)(out + plane_out + o));
    __builtin_nontemporal_store(hl, (v4f*)(out + 2 * plane_out + o));
    __builtin_nontemporal_store(hh, (v4f*)(out + 3 * plane_out + o));
}

extern "C" void kernel_launch(void* const* d_in, const int* in_sizes, int n_in,
                              void* d_out, int out_size, void* d_ws, size_t ws_size,
                              hipStream_t stream) {
    (void)n_in; (void)out_size; (void)d_ws; (void)ws_size;

    const float* x  = (const float*)d_in[0];
    float*      out = (float*)d_out;

    const long n_elem    = (long)in_sizes[0];          // B*C*H*W
    const int  n_img     = (int)(n_elem / (256 * 256)); // B*C images
    const long plane_out = (long)n_img * 128 * 128;

    const long total_threads = (long)n_img * 4096;      // 32 * 128 per image
    const int  block = 256;                             // 8 wave32 waves
    const int  grid  = (int)((total_threads + block - 1) / block);

    haar_dwt2_kernel<<<grid, block, 0, stream>>>(x, out, n_img, plane_out);
}